// SingleMemoryHop_69088843923725
// MI455X (gfx1250) — compile-verified
//
#include <hip/hip_runtime.h>

// CDNA5 / gfx1250, wave32. One block per batch (B=64), 1024 threads = 32 waves.
// Roofline: gather-bound (~671 MB of L2-resident embedding reads); matrix ops
// are negligible FLOPs, so WMMA is used only for the final partial reduction.

typedef __attribute__((ext_vector_type(2))) float v2f;
typedef __attribute__((ext_vector_type(8))) float v8f;

#define B_ 64
#define M_ 320
#define S_ 32
#define E_ 128

__global__ __launch_bounds__(1024) void SingleMemoryHop_kernel(
    const int* __restrict__ x,        // [B, M, S] (int32 per harness marshalling)
    const float* __restrict__ u,      // [B, 1, E]
    const float* __restrict__ mem_emb,// [V, E]
    const float* __restrict__ out_emb,// [V, E]
    float* __restrict__ out)          // [B, 1, E]
{
    __shared__ int   sidx[M_ * S_];     // 40 KB: staged index slab for this batch
    __shared__ float p_lds[M_];         // scores -> exp(score - max)
    __shared__ float part[32][E_];      // 16 KB: per-wave combine partials
    __shared__ float s_max, s_inv;

    const int b    = blockIdx.x;
    const int tid  = threadIdx.x;
    const int lane = tid & 31;
    const int wave = tid >> 5;

    // ---- Stage x[b,:,:] into LDS via CDNA5 async global->LDS loads (ASYNCcnt path)
    const int* xb = x + (size_t)b * (M_ * S_);
    for (int c = tid; c < (M_ * S_) / 4; c += 1024) {
        unsigned lds_off       = (unsigned)(size_t)(&sidx[c * 4]);
        unsigned long long ga  = (unsigned long long)(const void*)(xb + c * 4);
        asm volatile("global_load_async_to_lds_b128 %0, %1, off"
                     :: "v"(lds_off), "v"(ga) : "memory");
    }
    asm volatile("s_wait_asynccnt 0" ::: "memory");
    __syncthreads();

    // Each lane owns 4 embedding dims: lane -> e = lane*4 .. lane*4+3 (float4 = full 512B row per wave)
    const float4 uv = ((const float4*)(u + (size_t)b * E_))[lane];

    // ---- Phase 1: m[mi] = sum_s mem_emb[idx]; score[mi] = dot(u, m[mi])
    for (int mi = wave; mi < M_; mi += 32) {
        const int* ip = &sidx[mi * S_];
        float ax = 0.f, ay = 0.f, az = 0.f, aw = 0.f;
#pragma unroll
        for (int s = 0; s < S_; ++s) {
            const float4 r = ((const float4*)(mem_emb + (size_t)ip[s] * E_))[lane];
            ax += r.x; ay += r.y; az += r.z; aw += r.w;
        }
        float d = ax * uv.x + ay * uv.y + az * uv.z + aw * uv.w;
#pragma unroll
        for (int off = 16; off > 0; off >>= 1) d += __shfl_xor(d, off, 32);
        if (lane == 0) p_lds[mi] = d;
    }
    __syncthreads();

    // ---- Softmax over M=320 (deterministic wave-0 reductions; no float atomics)
    if (wave == 0) {
        float m = -3.402823466e38f;
        for (int i = lane; i < M_; i += 32) m = fmaxf(m, p_lds[i]);
#pragma unroll
        for (int off = 16; off > 0; off >>= 1) m = fmaxf(m, __shfl_xor(m, off, 32));
        if (lane == 0) s_max = m;
    }
    __syncthreads();
    {
        const float mx = s_max;
        for (int i = tid; i < M_; i += 1024) p_lds[i] = expf(p_lds[i] - mx);
    }
    __syncthreads();
    if (wave == 0) {
        float s = 0.f;
        for (int i = lane; i < M_; i += 32) s += p_lds[i];
#pragma unroll
        for (int off = 16; off > 0; off >>= 1) s += __shfl_xor(s, off, 32);
        if (lane == 0) s_inv = 1.0f / s;
    }
    __syncthreads();

    // ---- Phase 2: fused combine. o_partial += p[mi] * (sum_s out_emb[idx])
    float ox = 0.f, oy = 0.f, oz = 0.f, ow = 0.f;
    for (int mi = wave; mi < M_; mi += 32) {
        const int* ip = &sidx[mi * S_];
        float ax = 0.f, ay = 0.f, az = 0.f, aw = 0.f;
#pragma unroll
        for (int s = 0; s < S_; ++s) {
            const float4 r = ((const float4*)(out_emb + (size_t)ip[s] * E_))[lane];
            ax += r.x; ay += r.y; az += r.z; aw += r.w;
        }
        const float p = p_lds[mi];   // unnormalized exp; normalize once at the end
        ox += p * ax; oy += p * ay; oz += p * az; ow += p * aw;
    }
    ((float4*)part[wave])[lane] = make_float4(ox, oy, oz, ow);
    __syncthreads();

    // ---- Final reduction over the 32 wave-partials via WMMA:
    // o = ones(1x32) @ P(32x128). A = all-ones 16x4 (replicated), so the result
    // is invariant to K-slot permutations in the B-matrix lane layout.
    if (wave < 8) {
        const int n     = lane & 15;
        const int hi    = (lane >> 4) << 1;   // 0 for lanes 0-15, 2 for lanes 16-31
        const int ebase = wave * 16;
        v2f a; a.x = 1.0f; a.y = 1.0f;        // ones A-matrix (16x4 f32)
        v8f d = {};
#pragma unroll
        for (int kc = 0; kc < 8; ++kc) {      // K = 32 partial rows, 4 per WMMA
            const int k0 = kc * 4 + hi;
            v2f bb;
            bb.x = part[k0 + 0][ebase + n];
            bb.y = part[k0 + 1][ebase + n];
            d = __builtin_amdgcn_wmma_f32_16x16x4_f32(
                    /*neg_a=*/false, a, /*neg_b=*/false, bb,
                    /*c_mod=*/(short)0, d, /*reuse_a=*/false, /*reuse_b=*/false);
        }
        // D layout: VGPR0, lanes 0-15 -> M=0, N=lane (all M rows identical here)
        if (lane < 16) out[(size_t)b * E_ + ebase + lane] = d[0] * s_inv;
    }
}

extern "C" void kernel_launch(void* const* d_in, const int* in_sizes, int n_in,
                              void* d_out, int out_size, void* d_ws, size_t ws_size,
                              hipStream_t stream) {
    const int*   x   = (const int*)d_in[0];
    const float* u   = (const float*)d_in[1];
    const float* me  = (const float*)d_in[2];
    const float* oe  = (const float*)d_in[3];
    float*       out = (float*)d_out;
    (void)in_sizes; (void)n_in; (void)out_size; (void)d_ws; (void)ws_size;

    SingleMemoryHop_kernel<<<B_, 1024, 0, stream>>>(x, u, me, oe, out);
}